// APPNPModel_76149770158506
// MI455X (gfx1250) — compile-verified
//
#include <hip/hip_runtime.h>
#include <hip/hip_bf16.h>

// ---------------------------------------------------------------------------
// APPNP (k=1) on MI455X / gfx1250:
//   x   = relu(features @ W.T + b)            -> WMMA f32 16x16x4 GEMM
//   deg = segment counts over edges           -> global f32 atomics
//   agg = scatter_add(x[src]*rsqrt(outdeg))   -> float4 gather + f32 atomics (L2)
//   out = 0.8*agg*rsqrt(indeg) + 0.2*x        -> float4 streaming kernel
// ---------------------------------------------------------------------------

typedef float v2f __attribute__((ext_vector_type(2)));
typedef float v8f __attribute__((ext_vector_type(8)));

#define F_IN        512
#define C_OUT       16
#define LDSW_STRIDE (F_IN + 2)   // pad 2 floats/row -> staggered LDS banks

// ---------------------------------------------------------------------------
// GEMM + bias + relu via V_WMMA_F32_16X16X4_F32.
// One wave (32 lanes) computes one 16-row x 16-col output tile.
// A (16x4 f32): lane l in 0..15 holds A[l][k0+0..1], lane l+16 holds A[l][k0+2..3]
// B (4x16 f32): lane l in 0..15 holds B[k0+0..1][l], lane l+16 holds B[k0+2..3][l]
//   with B[k][n] = W[n][k]  (W is [16,512] row-major -> contiguous b64 per lane)
// D (16x16 f32, 8 VGPRs): VGPR r = row (r + 8*(lane>>4)), column = lane&15
// ---------------------------------------------------------------------------
__launch_bounds__(256)
__global__ void appnp_gemm_relu(const float* __restrict__ feat,
                                const float* __restrict__ W,
                                const float* __restrict__ bias,
                                float* __restrict__ xbuf,
                                int nTiles) {
    __shared__ float Wlds[C_OUT * LDSW_STRIDE];   // 16*514*4 = 32,896 B

    // Cooperative stage of W (16x512 f32 = 32 KB) into padded LDS.
    for (int i = threadIdx.x; i < (C_OUT * F_IN) / 4; i += blockDim.x) {
        const int   r  = (i * 4) / F_IN;
        const int   c  = (i * 4) % F_IN;
        const float4 w = ((const float4*)W)[i];
        float* dr = &Wlds[r * LDSW_STRIDE + c];
        dr[0] = w.x; dr[1] = w.y; dr[2] = w.z; dr[3] = w.w;
    }
    __syncthreads();

    const int wave = threadIdx.x >> 5;            // wave32
    const int lane = threadIdx.x & 31;
    const int tile = blockIdx.x * 8 + wave;
    if (tile >= nTiles) return;                   // whole wave exits -> EXEC ok

    const int half = lane >> 4;                   // 0: K={0,1}, 1: K={2,3}
    const int l    = lane & 15;

    const float* arow = feat + (size_t)(tile * 16 + l) * F_IN + 2 * half;
    const float* brow = &Wlds[l * LDSW_STRIDE + 2 * half];

    v8f acc = {};
#pragma unroll 8
    for (int k0 = 0; k0 < F_IN; k0 += 4) {
        v2f a = *(const v2f*)(arow + k0);         // global_load_b64
        v2f b = *(const v2f*)(brow + k0);         // ds_load_b64
        acc = __builtin_amdgcn_wmma_f32_16x16x4_f32(
            /*neg_a=*/false, a, /*neg_b=*/false, b,
            /*c_mod=*/(short)0, acc, /*reuse_a=*/false, /*reuse_b=*/false);
    }

    const float bv = bias[l];
#pragma unroll
    for (int r = 0; r < 8; ++r) {
        const int row = tile * 16 + r + 8 * half;
        float v = acc[r] + bv;
        v = v > 0.0f ? v : 0.0f;                  // relu
        xbuf[(size_t)row * C_OUT + l] = v;
    }
}

// ---------------------------------------------------------------------------
__global__ void appnp_zero(float* __restrict__ p, long long n) {
    long long i = (long long)blockIdx.x * blockDim.x + threadIdx.x;
    if (i < n) p[i] = 0.0f;
}

__global__ void appnp_degree(const int* __restrict__ src,
                             const int* __restrict__ dst,
                             float* __restrict__ outdeg,
                             float* __restrict__ indeg, int E) {
    int e = blockIdx.x * blockDim.x + threadIdx.x;
    if (e < E) {
        atomicAdd(&outdeg[src[e]], 1.0f);
        atomicAdd(&indeg[dst[e]], 1.0f);
    }
}

// 4 threads per edge; each handles a float4 (4 of 16 channels).
__global__ void appnp_scatter(const int* __restrict__ src,
                              const int* __restrict__ dst,
                              const float* __restrict__ xbuf,
                              const float* __restrict__ outdeg,
                              float* __restrict__ agg, int E) {
    long long t = (long long)blockIdx.x * blockDim.x + threadIdx.x;
    if (t >= (long long)E * 4) return;
    const int e = (int)(t >> 2);
    const int g = (int)(t & 3);
    const int s = src[e];
    const int d = dst[e];
    const float ns = rsqrtf(fmaxf(outdeg[s], 1.0f));
    const float4 v = ((const float4*)xbuf)[(size_t)s * 4 + g];
    float* ap = agg + (size_t)d * C_OUT + g * 4;
    atomicAdd(ap + 0, v.x * ns);
    atomicAdd(ap + 1, v.y * ns);
    atomicAdd(ap + 2, v.z * ns);
    atomicAdd(ap + 3, v.w * ns);
}

// out = 0.8 * agg * rsqrt(max(indeg,1)) + 0.2 * x
__global__ void appnp_finalize(const float* __restrict__ agg,
                               const float* __restrict__ indeg,
                               const float* __restrict__ xbuf,
                               float* __restrict__ out, int N) {
    int t = blockIdx.x * blockDim.x + threadIdx.x;
    if (t >= N * 4) return;
    const int n = t >> 2;
    const float nd = rsqrtf(fmaxf(indeg[n], 1.0f));
    const float4 a  = ((const float4*)agg)[t];
    const float4 x0 = ((const float4*)xbuf)[t];
    float4 o;
    o.x = 0.8f * a.x * nd + 0.2f * x0.x;
    o.y = 0.8f * a.y * nd + 0.2f * x0.y;
    o.z = 0.8f * a.z * nd + 0.2f * x0.z;
    o.w = 0.8f * a.w * nd + 0.2f * x0.w;
    ((float4*)out)[t] = o;
}

// ---------------------------------------------------------------------------
extern "C" void kernel_launch(void* const* d_in, const int* in_sizes, int n_in,
                              void* d_out, int out_size, void* d_ws, size_t ws_size,
                              hipStream_t stream) {
    const float* feat = (const float*)d_in[0];   // [N, 512]
    const float* W    = (const float*)d_in[1];   // [16, 512]
    const float* bias = (const float*)d_in[2];   // [16]
    const int*   src  = (const int*)d_in[3];     // [E]
    const int*   dst  = (const int*)d_in[4];     // [E]

    const int N = in_sizes[0] / F_IN;            // 100000
    const int E = in_sizes[3];                   // 3200000

    // Workspace layout (floats): xbuf[N*16] | agg[N*16] | outdeg[N] | indeg[N]
    float* ws     = (float*)d_ws;
    float* xbuf   = ws;
    float* agg    = xbuf + (size_t)N * C_OUT;
    float* outdeg = agg  + (size_t)N * C_OUT;
    float* indeg  = outdeg + N;

    // 1) zero agg + degree arrays (contiguous region), every call
    {
        long long cnt = (long long)N * C_OUT + 2LL * N;
        int blocks = (int)((cnt + 255) / 256);
        appnp_zero<<<blocks, 256, 0, stream>>>(agg, cnt);
    }
    // 2) degrees
    appnp_degree<<<(E + 255) / 256, 256, 0, stream>>>(src, dst, outdeg, indeg, E);
    // 3) WMMA GEMM + bias + relu
    {
        const int nTiles = (N + 15) / 16;        // 6250 (N divisible by 16)
        const int blocks = (nTiles + 7) / 8;     // 8 waves / block
        appnp_gemm_relu<<<blocks, 256, 0, stream>>>(feat, W, bias, xbuf, nTiles);
    }
    // 4) edge scatter-add
    {
        long long t = (long long)E * 4;
        int blocks = (int)((t + 255) / 256);
        appnp_scatter<<<blocks, 256, 0, stream>>>(src, dst, xbuf, outdeg, agg, E);
    }
    // 5) teleport mix
    appnp_finalize<<<(N * 4 + 255) / 256, 256, 0, stream>>>(agg, indeg, xbuf,
                                                            (float*)d_out, N);
}